// GraphAutoencoder_62672162783966
// MI455X (gfx1250) — compile-verified
//
#include <hip/hip_runtime.h>
#include <hip/hip_bf16.h>

#define IN_C 128
#define HID 32
#define HEADS 4
#define OUT_C 64
#define HC 128            // HEADS*HID
#define NEG_SLOPE 0.2f

typedef float v2f __attribute__((ext_vector_type(2)));
typedef float v8f __attribute__((ext_vector_type(8)));

// ---- monotonic float<->uint mapping for atomic max on floats (incl. negatives)
__device__ __forceinline__ unsigned f2ord(float f) {
    unsigned u = __float_as_uint(f);
    return (u >> 31) ? ~u : (u ^ 0x80000000u);
}
__device__ __forceinline__ float ord2f(unsigned o) {
    return __uint_as_float((o >> 31) ? (o ^ 0x80000000u) : ~o);
}

// ---------------------------------------------------------------- init
__global__ void k_init(float* agg, unsigned* emaxu, float* denom, int n) {
    int tid = blockIdx.x * blockDim.x + threadIdx.x;
    long long total = (long long)n * HC;
    if (tid < total) agg[tid] = 0.0f;
    if (tid < n * HEADS) { emaxu[tid] = 0u; denom[tid] = 0.0f; }
}

// ------------------------------------------- fused encoder GEMM (WMMA f32)
// xl = x@Wl + bl ; xr = x@Wr + br    x:[n,128] W:[128,128]
// one wave -> one 16x16 tile of both outputs; K marched 4 at a time.
__global__ __launch_bounds__(128) void k_gemm_enc(
    const float* __restrict__ x,
    const float* __restrict__ Wl, const float* __restrict__ bl,
    const float* __restrict__ Wr, const float* __restrict__ br,
    float* __restrict__ xl, float* __restrict__ xr, int n)
{
    const int numColTiles = HC / 16;                 // 8
    const int rowTiles = (n + 15) / 16;
    const int wave = blockIdx.x * 4 + threadIdx.y;
    if (wave >= rowTiles * numColTiles) return;
    const int rowTile = wave / numColTiles;
    const int colTile = wave % numColTiles;

    const int lane  = threadIdx.x;                   // 0..31 (wave32)
    const int lane15 = lane & 15;
    const int kh    = (lane >> 4) * 2;               // K offset pair select
    int row = rowTile * 16 + lane15;
    if (row >= n) row = n - 1;                       // clamp loads; stores guarded
    const int col = colTile * 16 + lane15;

    v8f accL = {0.f,0.f,0.f,0.f,0.f,0.f,0.f,0.f};
    v8f accR = {0.f,0.f,0.f,0.f,0.f,0.f,0.f,0.f};

    const float* xrow = x + (size_t)row * IN_C;
    #pragma unroll
    for (int k = 0; k < IN_C; k += 4) {
        v2f a;
        a.x = xrow[k + kh];
        a.y = xrow[k + kh + 1];
        v2f bL, bR;
        bL.x = Wl[(size_t)(k + kh) * HC + col];
        bL.y = Wl[(size_t)(k + kh + 1) * HC + col];
        bR.x = Wr[(size_t)(k + kh) * HC + col];
        bR.y = Wr[(size_t)(k + kh + 1) * HC + col];
        accL = __builtin_amdgcn_wmma_f32_16x16x4_f32(false, a, false, bL,
                                                     (short)0, accL, false, false);
        accR = __builtin_amdgcn_wmma_f32_16x16x4_f32(false, a, false, bR,
                                                     (short)0, accR, false, false);
    }

    const int mBase = rowTile * 16 + ((lane >> 4) * 8);
    const float biasL = bl[col];
    const float biasR = br[col];
    #pragma unroll
    for (int r = 0; r < 8; ++r) {
        const int m = mBase + r;
        if (m < n) {
            xl[(size_t)m * HC + col] = accL[r] + biasL;
            xr[(size_t)m * HC + col] = accR[r] + biasR;
        }
    }
}

// ------------------------------------------------- edge attention logits
// one thread per (edge, head)
__global__ void k_edge_logits(
    const float* __restrict__ xl, const float* __restrict__ xr,
    const long long* __restrict__ ei, const float* __restrict__ att,
    float* __restrict__ ebuf, unsigned* __restrict__ emaxu,
    int E, int Et)
{
    int tid = blockIdx.x * blockDim.x + threadIdx.x;
    if (tid >= Et * HEADS) return;
    const int edge = tid >> 2;
    const int h = tid & 3;
    long long src, dst;
    if (edge < E) { src = ei[edge]; dst = ei[E + edge]; }
    else          { src = dst = edge - E; }          // self loop

    const float* pl = xl + (size_t)src * HC + h * HID;
    const float* pr = xr + (size_t)dst * HC + h * HID;
    const float* pa = att + h * HID;
    float e = 0.0f;
    #pragma unroll
    for (int c = 0; c < HID; ++c) {
        float m = pl[c] + pr[c];
        m = (m > 0.0f) ? m : NEG_SLOPE * m;
        e += pa[c] * m;
    }
    ebuf[tid] = e;
    atomicMax(&emaxu[(size_t)dst * HEADS + h], f2ord(e));
}

// ------------------------------------------------- exp + denominator
__global__ void k_edge_exp(
    const long long* __restrict__ ei,
    float* __restrict__ ebuf, const unsigned* __restrict__ emaxu,
    float* __restrict__ denom, int E, int Et)
{
    int tid = blockIdx.x * blockDim.x + threadIdx.x;
    if (tid >= Et * HEADS) return;
    const int edge = tid >> 2;
    const int h = tid & 3;
    long long dst;
    if (edge < E) dst = ei[E + edge]; else dst = edge - E;

    const float mx = ord2f(emaxu[(size_t)dst * HEADS + h]);
    const float a = __expf(ebuf[tid] - mx);
    ebuf[tid] = a;
    atomicAdd(&denom[(size_t)dst * HEADS + h], a);
}

// ------------------------------------------------- weighted scatter
// one thread per (edge, channel) -> coalesced xl reads, L2-resident atomics
__global__ void k_edge_scatter(
    const long long* __restrict__ ei,
    const float* __restrict__ ebuf, const float* __restrict__ denom,
    const float* __restrict__ xl, float* __restrict__ agg,
    int E, int Et)
{
    long long tid = (long long)blockIdx.x * blockDim.x + threadIdx.x;
    if (tid >= (long long)Et * HC) return;
    const int edge = (int)(tid >> 7);
    const int ch = (int)(tid & (HC - 1));
    const int h = ch >> 5;
    long long src, dst;
    if (edge < E) { src = ei[edge]; dst = ei[E + edge]; }
    else          { src = dst = edge - E; }

    const float alpha = ebuf[(size_t)edge * HEADS + h] /
                        denom[(size_t)dst * HEADS + h];
    atomicAdd(&agg[(size_t)dst * HC + ch], alpha * xl[(size_t)src * HC + ch]);
}

// ------------------------------------------------- bias + ReLU (in place)
__global__ void k_encode(float* __restrict__ agg, const float* __restrict__ b_enc, int n) {
    long long tid = (long long)blockIdx.x * blockDim.x + threadIdx.x;
    if (tid >= (long long)n * HC) return;
    float v = agg[tid] + b_enc[(int)(tid & (HC - 1))];
    agg[tid] = (v > 0.0f) ? v : 0.0f;
}

// ------------------------------------------------- decoder GEMM (WMMA f32)
// dec = enc @ Wd + bd    enc:[n,128]  Wd:[128,64]
__global__ __launch_bounds__(128) void k_gemm_dec(
    const float* __restrict__ enc,
    const float* __restrict__ Wd, const float* __restrict__ bd,
    float* __restrict__ dec, int n)
{
    const int numColTiles = OUT_C / 16;              // 4
    const int rowTiles = (n + 15) / 16;
    const int wave = blockIdx.x * 4 + threadIdx.y;
    if (wave >= rowTiles * numColTiles) return;
    const int rowTile = wave / numColTiles;
    const int colTile = wave % numColTiles;

    const int lane = threadIdx.x;
    const int lane15 = lane & 15;
    const int kh = (lane >> 4) * 2;
    int row = rowTile * 16 + lane15;
    if (row >= n) row = n - 1;
    const int col = colTile * 16 + lane15;

    v8f acc = {0.f,0.f,0.f,0.f,0.f,0.f,0.f,0.f};
    const float* erow = enc + (size_t)row * HC;
    #pragma unroll
    for (int k = 0; k < HC; k += 4) {
        v2f a;
        a.x = erow[k + kh];
        a.y = erow[k + kh + 1];
        v2f b;
        b.x = Wd[(size_t)(k + kh) * OUT_C + col];
        b.y = Wd[(size_t)(k + kh + 1) * OUT_C + col];
        acc = __builtin_amdgcn_wmma_f32_16x16x4_f32(false, a, false, b,
                                                    (short)0, acc, false, false);
    }

    const int mBase = rowTile * 16 + ((lane >> 4) * 8);
    const float bias = bd[col];
    #pragma unroll
    for (int r = 0; r < 8; ++r) {
        const int m = mBase + r;
        if (m < n) dec[(size_t)m * OUT_C + col] = acc[r] + bias;
    }
}

// ================================================================ launch
extern "C" void kernel_launch(void* const* d_in, const int* in_sizes, int n_in,
                              void* d_out, int out_size, void* d_ws, size_t ws_size,
                              hipStream_t stream) {
    const float*     x    = (const float*)d_in[0];
    const long long* ei   = (const long long*)d_in[1];   // int64 [2,E]
    const float*     Wl   = (const float*)d_in[2];
    const float*     bl   = (const float*)d_in[3];
    const float*     Wr   = (const float*)d_in[4];
    const float*     br   = (const float*)d_in[5];
    const float*     att  = (const float*)d_in[6];
    const float*     benc = (const float*)d_in[7];
    const float*     Wd   = (const float*)d_in[8];
    const float*     bd   = (const float*)d_in[9];

    const int N  = in_sizes[0] / IN_C;
    const int E  = in_sizes[1] / 2;
    const int Et = E + N;                                 // + self loops

    float* out_enc = (float*)d_out;                       // [N,128]
    float* out_dec = out_enc + (size_t)N * HC;            // [N,64]

    // workspace layout
    float* xl = (float*)d_ws;
    float* xr = xl + (size_t)N * HC;
    float* ebuf = xr + (size_t)N * HC;                    // [Et,4]
    unsigned* emaxu = (unsigned*)(ebuf + (size_t)Et * HEADS);
    float* denom = (float*)(emaxu + (size_t)N * HEADS);

    const int TB = 256;

    // 1) init aggregate (encoded region of d_out) + emax/denom
    {
        long long total = (long long)N * HC;
        int blocks = (int)((total + TB - 1) / TB);
        k_init<<<blocks, TB, 0, stream>>>(out_enc, emaxu, denom, N);
    }
    // 2) encoder GEMMs (WMMA)
    {
        int rowTiles = (N + 15) / 16;
        int waves = rowTiles * (HC / 16);
        dim3 blk(32, 4);
        k_gemm_enc<<<(waves + 3) / 4, blk, 0, stream>>>(x, Wl, bl, Wr, br, xl, xr, N);
    }
    // 3) attention logits + segment max
    {
        int total = Et * HEADS;
        k_edge_logits<<<(total + TB - 1) / TB, TB, 0, stream>>>(xl, xr, ei, att,
                                                                ebuf, emaxu, E, Et);
    }
    // 4) exp + segment sum
    {
        int total = Et * HEADS;
        k_edge_exp<<<(total + TB - 1) / TB, TB, 0, stream>>>(ei, ebuf, emaxu,
                                                             denom, E, Et);
    }
    // 5) weighted scatter into encoded region
    {
        long long total = (long long)Et * HC;
        int blocks = (int)((total + TB - 1) / TB);
        k_edge_scatter<<<blocks, TB, 0, stream>>>(ei, ebuf, denom, xl, out_enc, E, Et);
    }
    // 6) bias + ReLU (in place -> encoded output)
    {
        long long total = (long long)N * HC;
        int blocks = (int)((total + TB - 1) / TB);
        k_encode<<<blocks, TB, 0, stream>>>(out_enc, benc, N);
    }
    // 7) decoder GEMM (WMMA)
    {
        int rowTiles = (N + 15) / 16;
        int waves = rowTiles * (OUT_C / 16);
        dim3 blk(32, 4);
        k_gemm_dec<<<(waves + 3) / 4, blk, 0, stream>>>(out_enc, Wd, bd, out_dec, N);
    }
}